// GCN_61375082659924
// MI455X (gfx1250) — compile-verified
//
#include <hip/hip_runtime.h>
#include <math.h>

#define N_NODES 200000
#define N_EDGES 3200000
#define N_FEAT  602
#define N_CLASS 41
#define KITER   151   // ceil(602/4)
#define NTILES  3     // ceil(41/16)

typedef float v2f __attribute__((ext_vector_type(2)));
typedef float v8f __attribute__((ext_vector_type(8)));

// ---------------- degree / normalization ----------------
__global__ void k_init_deg(float* __restrict__ deg) {
    int i = blockIdx.x * 256 + threadIdx.x;
    if (i < N_NODES) deg[i] = 1.0f;   // self loop contributes 1
}

__global__ void k_deg(const long long* __restrict__ ei, float* __restrict__ deg) {
    int e = blockIdx.x * 256 + threadIdx.x;
    if (e < N_EDGES) {
        long long d = ei[(size_t)N_EDGES + e];   // dst row
        atomicAdd(&deg[d], 1.0f);
    }
}

__global__ void k_dinv(float* __restrict__ deg) {
    int i = blockIdx.x * 256 + threadIdx.x;
    if (i < N_NODES) deg[i] = rsqrtf(deg[i]);    // deg >= 1 always
}

// ---------------- pack W into WMMA B-operand lane layout ----------------
// For iteration it (K base = 4*it), tile t, lane l:
//   half = l>>4, n = t*16 + (l&15), k0 = 4*it + 2*half
//   bpack float2 = { W[k0][n], W[k0+1][n] }   (zero padded OOB)
__global__ void k_packb(const float* __restrict__ W, float* __restrict__ bpack) {
    int idx = blockIdx.x * 256 + threadIdx.x;
    if (idx >= KITER * NTILES * 32) return;
    int lane = idx & 31;
    int t    = (idx >> 5) % NTILES;
    int it   = idx / (32 * NTILES);
    int half = lane >> 4;
    int n    = t * 16 + (lane & 15);
    int k0   = it * 4 + 2 * half;
    float b0 = (k0     < N_FEAT && n < N_CLASS) ? W[k0 * N_CLASS + n]       : 0.0f;
    float b1 = (k0 + 1 < N_FEAT && n < N_CLASS) ? W[(k0 + 1) * N_CLASS + n] : 0.0f;
    bpack[2 * idx]     = b0;
    bpack[2 * idx + 1] = b1;
}

// ---------------- WMMA GEMM: h = x @ W ----------------
__global__ void __launch_bounds__(256) k_gemm(const float* __restrict__ x,
                                              const float* __restrict__ bpack,
                                              float* __restrict__ h) {
    int wave  = threadIdx.x >> 5;
    int lane  = threadIdx.x & 31;
    int mtile = blockIdx.x * 8 + wave;
    if (mtile >= (N_NODES / 16)) return;   // uniform per wave -> EXEC stays all-1s

    int half = lane >> 4;
    int mrow = lane & 15;
    const float* xrow = x + (size_t)(mtile * 16 + mrow) * N_FEAT + 2 * half;
    const v2f* bp = (const v2f*)bpack;

    v8f acc0 = {}; v8f acc1 = {}; v8f acc2 = {};

    for (int it = 0; it < KITER - 1; ++it) {
        v2f a  = *(const v2f*)(xrow + it * 4);            // x[m][4it+2h .. +1], 8B aligned
        v2f b0 = bp[(it * NTILES + 0) * 32 + lane];
        v2f b1 = bp[(it * NTILES + 1) * 32 + lane];
        v2f b2 = bp[(it * NTILES + 2) * 32 + lane];
        acc0 = __builtin_amdgcn_wmma_f32_16x16x4_f32(false, a, false, b0, (short)0, acc0, false, false);
        acc1 = __builtin_amdgcn_wmma_f32_16x16x4_f32(false, a, false, b1, (short)0, acc1, false, false);
        acc2 = __builtin_amdgcn_wmma_f32_16x16x4_f32(false, a, false, b2, (short)0, acc2, false, false);
    }
    {   // epilogue it=150: K=600,601 valid (half 0); K=602,603 OOB (half 1) -> zeros
        const int it = KITER - 1;
        v2f a;
        if (half == 0) a = *(const v2f*)(xrow + it * 4);
        else { a.x = 0.0f; a.y = 0.0f; }
        v2f b0 = bp[(it * NTILES + 0) * 32 + lane];
        v2f b1 = bp[(it * NTILES + 1) * 32 + lane];
        v2f b2 = bp[(it * NTILES + 2) * 32 + lane];
        acc0 = __builtin_amdgcn_wmma_f32_16x16x4_f32(false, a, false, b0, (short)0, acc0, false, false);
        acc1 = __builtin_amdgcn_wmma_f32_16x16x4_f32(false, a, false, b1, (short)0, acc1, false, false);
        acc2 = __builtin_amdgcn_wmma_f32_16x16x4_f32(false, a, false, b2, (short)0, acc2, false, false);
    }

    // C/D layout: VGPR v, lane l -> row 8*half+v, col l&15
#pragma unroll
    for (int v = 0; v < 8; ++v) {
        int row = mtile * 16 + half * 8 + v;
        float* hrow = h + (size_t)row * N_CLASS;
        hrow[mrow]      = acc0[v];            // cols 0..15
        hrow[16 + mrow] = acc1[v];            // cols 16..31
        if (32 + mrow < N_CLASS) hrow[32 + mrow] = acc2[v];   // cols 32..40
    }
}

// ---------------- self-loop + bias init: agg = b + h*dinv^2 ----------------
__global__ void k_init_agg(const float* __restrict__ h, const float* __restrict__ dinv,
                           const float* __restrict__ bias, float* __restrict__ agg) {
    int idx = blockIdx.x * 256 + threadIdx.x;
    if (idx >= N_NODES * N_CLASS) return;
    int m = idx / N_CLASS;
    int c = idx - m * N_CLASS;
    float di = dinv[m];
    agg[idx] = bias[c] + h[idx] * di * di;
}

// ---------------- edge scatter-add (8 threads per edge) ----------------
__global__ void k_edge(const long long* __restrict__ ei, const float* __restrict__ h,
                       const float* __restrict__ dinv, float* __restrict__ agg) {
    int tid = blockIdx.x * 256 + threadIdx.x;
    int e   = tid >> 3;
    if (e >= N_EDGES) return;
    int sub = tid & 7;
    long long s = ei[e];
    long long d = ei[(size_t)N_EDGES + e];
    float norm = dinv[s] * dinv[d];
    const float* hs = h + (size_t)s * N_CLASS;
    float* ad = agg + (size_t)d * N_CLASS;
    for (int c = sub; c < N_CLASS; c += 8)
        atomicAdd(&ad[c], hs[c] * norm);
}

// ---------------- row-wise log_softmax: one wave32 per row ----------------
__global__ void k_lsm(const float* __restrict__ agg, float* __restrict__ z) {
    int wave = threadIdx.x >> 5;
    int lane = threadIdx.x & 31;
    int row  = blockIdx.x * 8 + wave;
    if (row >= N_NODES) return;
    const float* a = agg + (size_t)row * N_CLASS;
    float v0 = a[lane];                                        // lane < 32 <= 41, valid
    bool  hi = (lane + 32) < N_CLASS;
    float v1 = hi ? a[lane + 32] : -INFINITY;
    float m = fmaxf(v0, v1);
    for (int o = 16; o > 0; o >>= 1) m = fmaxf(m, __shfl_xor(m, o, 32));
    float sum = expf(v0 - m) + (hi ? expf(v1 - m) : 0.0f);
    for (int o = 16; o > 0; o >>= 1) sum += __shfl_xor(sum, o, 32);
    float lse = m + logf(sum);
    float* zr = z + (size_t)row * N_CLASS;
    zr[lane] = v0 - lse;
    if (hi) zr[lane + 32] = v1 - lse;
}

extern "C" void kernel_launch(void* const* d_in, const int* in_sizes, int n_in,
                              void* d_out, int out_size, void* d_ws, size_t ws_size,
                              hipStream_t stream) {
    const float*     x  = (const float*)d_in[0];
    const float*     W  = (const float*)d_in[1];
    const float*     b  = (const float*)d_in[2];
    const long long* ei = (const long long*)d_in[3];   // int64 [2][E]

    float* agg = (float*)d_out;                        // output 0
    float* z   = agg + (size_t)N_NODES * N_CLASS;      // output 1
    float* h   = z;                                    // stage h in z region (dead until k_lsm)

    float* deg   = (float*)d_ws;                       // N_NODES floats (deg -> dinv in place)
    float* bpack = deg + N_NODES;                      // KITER*NTILES*32 float2 (116 KB)

    k_init_deg<<<(N_NODES + 255) / 256, 256, 0, stream>>>(deg);
    k_deg     <<<(N_EDGES + 255) / 256, 256, 0, stream>>>(ei, deg);
    k_dinv    <<<(N_NODES + 255) / 256, 256, 0, stream>>>(deg);
    k_packb   <<<(KITER * NTILES * 32 + 255) / 256, 256, 0, stream>>>(W, bpack);
    k_gemm    <<<(N_NODES / 16 + 7) / 8, 256, 0, stream>>>(x, bpack, h);
    k_init_agg<<<(N_NODES * N_CLASS + 255) / 256, 256, 0, stream>>>(h, deg, b, agg);
    k_edge    <<<(N_EDGES * 8 + 255) / 256, 256, 0, stream>>>(ei, h, deg, agg);
    k_lsm     <<<(N_NODES + 7) / 8, 256, 0, stream>>>(agg, z);
}